// EMATrend_12352325943307
// MI455X (gfx1250) — compile-verified
//
#include <hip/hip_runtime.h>

typedef __attribute__((ext_vector_type(2))) float v2f;
typedef __attribute__((ext_vector_type(4))) float v4f;
typedef __attribute__((ext_vector_type(8))) float v8f;

#define GAM 0.99f
#define OMG 0.01f            // 1 - GAMMA

// Problem geometry (from reference): x is (32, 4096, 512) f32
#define BB   32
#define TT   4096
#define DD   512
#define CL   256             // chunk length along T
#define NCHUNK 16            // TT / CL
#define NTILE  16            // CL / 16 time-tiles per chunk
#define CTILES 32            // DD / 16 channel tiles

static constexpr size_t NTOT = (size_t)BB * TT * DD;   // 67,108,864

__device__ __forceinline__ float gpow_rt(int e) {
    float p = 1.f;
    for (int q = 0; q < e; ++q) p *= GAM;
    return p;
}

// ---------------------------------------------------------------------------
// Pass 1: per-(channel, chunk) local-scan terminal value P.
// trend[(i+1)*CL-1] = g^CL * trend[i*CL-1] + P_i
// ---------------------------------------------------------------------------
__global__ __launch_bounds__(256)
void ema_pass1(const float* __restrict__ x, float* __restrict__ P) {
    int tid   = blockIdx.x * 256 + threadIdx.x;
    int d     = tid & (DD - 1);
    int chunk = (tid >> 9) & (NCHUNK - 1);
    int b     = tid >> 13;

    const float* p0 = x + ((size_t)(b * TT + chunk * CL) * DD + d);
    float acc;
    if (chunk == 0) {
        acc = p0[0];                       // trend[0] = x[0]
        #pragma unroll 5
        for (int k = 1; k < CL; ++k)
            acc = fmaf(OMG, p0[(size_t)k * DD], GAM * acc);
    } else {
        acc = 0.f;                         // local scan with zero carry-in
        #pragma unroll 8
        for (int k = 0; k < CL; ++k)
            acc = fmaf(OMG, p0[(size_t)k * DD], GAM * acc);
    }
    P[((size_t)(b * DD + d) << 4) + chunk] = acc;
}

// ---------------------------------------------------------------------------
// Pass 2: WMMA-tiled scan. One wave per (b, channel-tile, chunk).
// trend    = W * x_tile     + g^{t+1}*carry   (W lower-triangular EMA matrix)
// residual = (I-W) * x_tile - g^{t+1}*carry   (second WMMA chain; no LDS
//                                              readback of x at D layout)
// All WMMAs use C=0 and sit off the serial carry chain. No block barriers
// (single-wave WG, LDS in-order per wave). Nontemporal streaming stores.
// ---------------------------------------------------------------------------
__global__ __launch_bounds__(32)
void ema_pass2(const float* __restrict__ x, const float* __restrict__ P,
               float* __restrict__ out) {
    int wid   = blockIdx.x;
    int chunk = wid & (NCHUNK - 1);
    int ct    = (wid >> 4) & (CTILES - 1);
    int b     = wid >> 9;

    int lane = threadIdx.x;
    int hi   = lane >> 4;        // half-wave: selects K/M groups per ISA layout
    int n    = lane & 15;        // column (channel) index within half
    int d0   = ct << 4;

    __shared__ float tile[2][16 * 20];   // double-buffered, 80B padded rows

    // Carry coefficients: g^{M+1}, M = r + 8*hi (and negated for residual)
    float cvec[8], ncvec[8];
    {
        float g8  = gpow_rt(8);
        float hiM = hi ? g8 : 1.f;
        float gp  = GAM;                  // g^{r+1}
        #pragma unroll
        for (int r = 0; r < 8; ++r) {
            cvec[r] = gp * hiM; ncvec[r] = -cvec[r]; gp *= GAM;
        }
    }

    // g^CL by 8 squarings
    float gcl = GAM;
    for (int q = 0; q < 8; ++q) gcl *= gcl;

    // A fragments, 16x4 f32 A layout: lanes0-15 hold K={k0,k0+1},
    // lanes16-31 K={k0+2,k0+3}.
    //   W[m][k] = (1-g) g^{m-k} (m>=k), 0 above diag        -> trend
    //   R[m][k] = I - W: diag g, -(1-g) g^{m-k} below, 0 up -> residual
    v2f afW[4], afR[4];
    #pragma unroll
    for (int kk = 0; kk < 4; ++kk) {
        int k0 = kk * 4 + hi * 2;
        int e0 = n - k0, e1 = e0 - 1;
        float w0 = (e0 >= 0) ? OMG * gpow_rt(e0) : 0.f;
        float w1 = (e1 >= 0) ? OMG * gpow_rt(e1) : 0.f;
        afW[kk].x = w0;
        afW[kk].y = w1;
        afR[kk].x = (e0 == 0) ? 1.f - w0 : -w0;
        afR[kk].y = (e1 == 0) ? 1.f - w1 : -w1;
    }
    // First-tile variants: column 0 of W zeroed (x[0] enters via carry);
    // column 0 of R becomes e0 (row0 = 1, rest 0).
    v2f aW0 = afW[0], aR0 = afR[0];
    if (!hi) { aW0.x = 0.f; aR0.x = (n == 0) ? 1.f : 0.f; }

    // Initial carry = trend[chunk*CL - 1]; chunk 0 uses carry' = x[0]/g
    float carry;
    if (chunk == 0) {
        carry = x[(size_t)b * TT * DD + d0 + n] * (1.f / GAM);
    } else {
        carry = 0.f;
        const float* Pc = P + ((size_t)(b * DD + d0 + n) << 4);
        for (int j = 0; j < chunk; ++j) carry = fmaf(gcl, carry, Pc[j]);
    }

    // Incremental base pointers (all per-tile offsets become immediates)
    const size_t chanBase = (size_t)b * TT * DD + d0;
    const int lr = lane >> 2;            // 0..7   (staging row)
    const int lc = (lane & 3) << 2;      // 0,4,8,12 (staging col, float4)
    const float* xload = x + chanBase + (size_t)(chunk * CL + lr) * DD + lc;
    float* o0 = out + chanBase + (size_t)(chunk * CL + hi * 8) * DD + n;
    float* o1 = o0 + NTOT;

    // Prime the pipeline: tile 0 in registers
    v4f c0 = *(const v4f*)xload;
    v4f c1 = *(const v4f*)(xload + (size_t)8 * DD);

    for (int tt = 0; tt < NTILE; ++tt) {
        const int buf = tt & 1;
        // Single-wave workgroup: DS ops are in-order per wave; compiler
        // preserves program order via aliasing on `tile`.
        *(v4f*)&tile[buf][lr * 20 + lc]       = c0;
        *(v4f*)&tile[buf][(lr + 8) * 20 + lc] = c1;
        if (tt + 1 < NTILE) {            // issue next tile's loads early;
            xload += (size_t)16 * DD;    // they overlap WMMA + stores below
            c0 = *(const v4f*)xload;
            c1 = *(const v4f*)(xload + (size_t)8 * DD);
        }

        // Two independent K=16 triangular matmuls (C = 0 inline)
        v8f accW = {}, accR = {};
        const bool first = (chunk == 0) && (tt == 0);
        #pragma unroll
        for (int kk = 0; kk < 4; ++kk) {
            v2f aw = afW[kk], ar = afR[kk];
            if (kk == 0 && first) { aw = aW0; ar = aR0; }
            int k0 = kk * 4 + hi * 2;    // B rows mirror A's K grouping
            v2f bf;
            bf.x = tile[buf][k0 * 20 + n];
            bf.y = tile[buf][(k0 + 1) * 20 + n];
            accW = __builtin_amdgcn_wmma_f32_16x16x4_f32(
                false, aw, false, bf, (short)0, accW, false, false);
            accR = __builtin_amdgcn_wmma_f32_16x16x4_f32(
                false, ar, false, bf, (short)0, accR, false, false);
        }

        // Apply carry to both planes, then store in two clauses.
        float tr[8], rs[8];
        #pragma unroll
        for (int r = 0; r < 8; ++r) {
            tr[r] = fmaf(cvec[r],  carry, accW[r]);
            rs[r] = fmaf(ncvec[r], carry, accR[r]);
        }
        #pragma unroll
        for (int r = 0; r < 8; ++r)
            __builtin_nontemporal_store(rs[r], &o0[r * DD]);  // residual
        #pragma unroll
        for (int r = 0; r < 8; ++r)
            __builtin_nontemporal_store(tr[r], &o1[r * DD]);  // trend
        o0 += (size_t)16 * DD;
        o1 += (size_t)16 * DD;

        // carry <- trend[row 15], held by lanes 16..31 in tr[7]
        carry = __shfl(tr[7], 16 + n, 32);
    }
}

// ---------------------------------------------------------------------------
extern "C" void kernel_launch(void* const* d_in, const int* in_sizes, int n_in,
                              void* d_out, int out_size, void* d_ws, size_t ws_size,
                              hipStream_t stream) {
    const float* x = (const float*)d_in[0];
    float* out = (float*)d_out;
    float* P   = (float*)d_ws;   // needs 32*512*16*4 B = 1 MiB

    ema_pass1<<<(BB * NCHUNK * DD) / 256, 256, 0, stream>>>(x, P);
    ema_pass2<<<BB * CTILES * NCHUNK, 32, 0, stream>>>(x, P, out);
}